// GTLayer_17901423690016
// MI455X (gfx1250) — compile-verified
//
#include <hip/hip_runtime.h>

#define H 64
#define LN_EPS 1e-6f

typedef __attribute__((ext_vector_type(2))) float v2f;
typedef __attribute__((ext_vector_type(8))) float v8f;

// D = A(16x4) * B(4x16) + C(16x16), fp32, wave32.
// A layout: lanes 0-15 hold row M=lane, VGPR0=K0,VGPR1=K1; lanes 16-31 hold K2,K3.
// B layout mirrors A: lanes 0-15 hold col N=lane with K0/K1, lanes 16-31 with K2/K3.
__device__ __forceinline__ v8f wmma4(v2f a, v2f b, v8f c) {
  return __builtin_amdgcn_wmma_f32_16x16x4_f32(false, a, false, b, (short)0, c,
                                               false, false);
}

__global__ void zero_kernel(float* __restrict__ p, size_t n) {
  size_t i = (size_t)blockIdx.x * blockDim.x + threadIdx.x;
  size_t stride = (size_t)gridDim.x * blockDim.x;
  for (; i < n; i += stride) p[i] = 0.0f;
}

// Stage weight W[KDIM x 64] into LDS in K-pair interleaved layout:
//   sW[(k>>1)*128 + col*2 + (k&1)] = W[k*64 + col]
// so a WMMA B-fragment {W[k][col], W[k+1][col]} is one aligned ds_load_b64.
template <int KDIM>
__device__ __forceinline__ void stage_weights(float* sW,
                                              const float* __restrict__ W,
                                              int tid) {
  for (int i = tid; i < KDIM * H; i += 256) {
    const int k = i >> 6;
    const int col = i & 63;
    sW[(k >> 1) * (2 * H) + col * 2 + (k & 1)] = W[i];
  }
}

// ---------------------------------------------------------------------------
// Edge kernel: new_e = [edge | node[snd] | node[rcv]] (1x192) @ We (192x64) + be
//   edges_out = LN(new_e + edge);  recv_agg[rcv] += new_e  (atomic, L2-resident)
// One wave per 16 edges; 8 waves per block; We staged in LDS (paired layout).
// ---------------------------------------------------------------------------
__global__ __launch_bounds__(256) void edge_kernel(
    const float* __restrict__ nodes, const float* __restrict__ edges,
    const int* __restrict__ senders, const int* __restrict__ receivers,
    const float* __restrict__ We, const float* __restrict__ be,
    const float* __restrict__ ln_s, const float* __restrict__ ln_b,
    float* __restrict__ edges_out, float* __restrict__ recv_agg, int nEdges) {
  extern __shared__ float lds[];
  float* sW = lds;                 // 192*64 floats = 48 KB (paired layout)
  float* sT = lds + 192 * H;       // 8 waves * 16*64 floats = 32 KB

  const int tid = threadIdx.x;
  stage_weights<192>(sW, We, tid);
  __syncthreads();

  const int wave = tid >> 5;
  const int lane = tid & 31;
  const int tile = blockIdx.x * 8 + wave;
  if (tile * 16 >= nEdges) return;  // wave-uniform exit

  const int mrow = lane & 15;            // row within 16-edge tile
  const int kk   = (lane >> 4) * 2;      // K sub-offset for this half-wave
  const int cc   = lane & 15;            // column within 16-wide N tile
  const int row  = tile * 16 + mrow;

  const float* rowE = edges + (size_t)row * H;
  const float* rowS = nodes + (size_t)senders[row] * H;
  const float* rowR = nodes + (size_t)receivers[row] * H;
  const float* const segs[3] = {rowE, rowS, rowR};

  v8f acc[4];
#pragma unroll
  for (int t = 0; t < 4; ++t) acc[t] = (v8f){0, 0, 0, 0, 0, 0, 0, 0};

#pragma unroll
  for (int s = 0; s < 3; ++s) {
    const float* P = segs[s];
    const int kbase = s * H;
#pragma unroll
    for (int k0 = 0; k0 < H; k0 += 4) {
      v2f a = *(const v2f*)(P + k0 + kk);
      // K-pair row for K = kbase+k0+kk (kk in {0,2} -> pair index +0/+1)
      const float* bp = sW + ((kbase + k0 + kk) >> 1) * (2 * H);
#pragma unroll
      for (int t = 0; t < 4; ++t) {
        v2f b = *(const v2f*)(bp + (t * 16 + cc) * 2);  // one ds_load_b64
        acc[t] = wmma4(a, b, acc[t]);
      }
    }
  }

  // Spill 16x64 tile (with bias) to this wave's LDS region.
  float* tileM = sT + wave * 16 * H;
  const int cm = (lane >> 4) * 8;  // C layout: VGPR r -> M=r (+8 for hi half)
#pragma unroll
  for (int t = 0; t < 4; ++t)
#pragma unroll
    for (int r = 0; r < 8; ++r)
      tileM[(cm + r) * H + t * 16 + cc] = acc[t][r] + be[t * 16 + cc];
  asm volatile("s_wait_dscnt 0" ::: "memory");
  __builtin_amdgcn_wave_barrier();

  // Residual + LayerNorm: 2 lanes per row, 32 columns each.
  const int halfc = (lane >> 4) * 32;
  const float* eRow = edges + (size_t)(tile * 16 + mrow) * H;
  float s1 = 0.0f, s2 = 0.0f;
#pragma unroll
  for (int c = 0; c < 32; ++c) {
    float v = tileM[mrow * H + halfc + c] + eRow[halfc + c];
    s1 += v;
    s2 += v * v;
  }
  s1 += __shfl_xor(s1, 16, 32);
  s2 += __shfl_xor(s2, 16, 32);
  const float mu = s1 * (1.0f / H);
  const float var = s2 * (1.0f / H) - mu * mu;
  const float rstd = rsqrtf(var + LN_EPS);

  float* oRow = edges_out + (size_t)(tile * 16 + mrow) * H;
  float* agg = recv_agg + (size_t)receivers[tile * 16 + mrow] * H;
#pragma unroll
  for (int c = 0; c < 32; ++c) {
    const int col = halfc + c;
    const float ne = tileM[mrow * H + col];          // new_edge (pre-residual)
    const float v = ne + eRow[col];
    oRow[col] = (v - mu) * rstd * ln_s[col] + ln_b[col];
    atomicAdd(agg + col, ne);                        // segment_sum into L2
  }
}

// ---------------------------------------------------------------------------
// Node kernel: new_n = [node | recv_agg] (1x128) @ Wn (128x64) + bn
//   nodes_out = LN(new_n + node)
// ---------------------------------------------------------------------------
__global__ __launch_bounds__(256) void node_kernel(
    const float* __restrict__ nodes, const float* __restrict__ recv_agg,
    const float* __restrict__ Wn, const float* __restrict__ bn,
    const float* __restrict__ ln_s, const float* __restrict__ ln_b,
    float* __restrict__ nodes_out, int nNodes) {
  extern __shared__ float lds[];
  float* sW = lds;                 // 128*64 floats = 32 KB (paired layout)
  float* sT = lds + 128 * H;       // 8 waves * 16*64 floats = 32 KB

  const int tid = threadIdx.x;
  stage_weights<128>(sW, Wn, tid);
  __syncthreads();

  const int wave = tid >> 5;
  const int lane = tid & 31;
  const int tile = blockIdx.x * 8 + wave;
  if (tile * 16 >= nNodes) return;  // wave-uniform exit

  const int mrow = lane & 15;
  const int kk   = (lane >> 4) * 2;
  const int cc   = lane & 15;
  const int row  = tile * 16 + mrow;

  const float* rowN = nodes + (size_t)row * H;
  const float* rowA = recv_agg + (size_t)row * H;
  const float* const segs[2] = {rowN, rowA};

  v8f acc[4];
#pragma unroll
  for (int t = 0; t < 4; ++t) acc[t] = (v8f){0, 0, 0, 0, 0, 0, 0, 0};

#pragma unroll
  for (int s = 0; s < 2; ++s) {
    const float* P = segs[s];
    const int kbase = s * H;
#pragma unroll
    for (int k0 = 0; k0 < H; k0 += 4) {
      v2f a = *(const v2f*)(P + k0 + kk);
      const float* bp = sW + ((kbase + k0 + kk) >> 1) * (2 * H);
#pragma unroll
      for (int t = 0; t < 4; ++t) {
        v2f b = *(const v2f*)(bp + (t * 16 + cc) * 2);
        acc[t] = wmma4(a, b, acc[t]);
      }
    }
  }

  float* tileM = sT + wave * 16 * H;
  const int cm = (lane >> 4) * 8;
#pragma unroll
  for (int t = 0; t < 4; ++t)
#pragma unroll
    for (int r = 0; r < 8; ++r)
      tileM[(cm + r) * H + t * 16 + cc] = acc[t][r] + bn[t * 16 + cc];
  asm volatile("s_wait_dscnt 0" ::: "memory");
  __builtin_amdgcn_wave_barrier();

  const int halfc = (lane >> 4) * 32;
  const float* nRow = nodes + (size_t)(tile * 16 + mrow) * H;
  float s1 = 0.0f, s2 = 0.0f;
#pragma unroll
  for (int c = 0; c < 32; ++c) {
    float v = tileM[mrow * H + halfc + c] + nRow[halfc + c];
    s1 += v;
    s2 += v * v;
  }
  s1 += __shfl_xor(s1, 16, 32);
  s2 += __shfl_xor(s2, 16, 32);
  const float mu = s1 * (1.0f / H);
  const float var = s2 * (1.0f / H) - mu * mu;
  const float rstd = rsqrtf(var + LN_EPS);

  float* oRow = nodes_out + (size_t)(tile * 16 + mrow) * H;
#pragma unroll
  for (int c = 0; c < 32; ++c) {
    const int col = halfc + c;
    const float v = tileM[mrow * H + col] + nRow[col];
    oRow[col] = (v - mu) * rstd * ln_s[col] + ln_b[col];
  }
}

extern "C" void kernel_launch(void* const* d_in, const int* in_sizes, int n_in,
                              void* d_out, int out_size, void* d_ws,
                              size_t ws_size, hipStream_t stream) {
  const float* nodes      = (const float*)d_in[0];
  const float* edges      = (const float*)d_in[1];
  const int*   senders    = (const int*)d_in[2];
  const int*   receivers  = (const int*)d_in[3];
  const float* We         = (const float*)d_in[4];
  const float* be         = (const float*)d_in[5];
  const float* Wn         = (const float*)d_in[6];
  const float* bn         = (const float*)d_in[7];
  const float* ln_n_scale = (const float*)d_in[8];
  const float* ln_n_bias  = (const float*)d_in[9];
  const float* ln_e_scale = (const float*)d_in[10];
  const float* ln_e_bias  = (const float*)d_in[11];

  const int nNodes = in_sizes[0] / H;
  const int nEdges = in_sizes[2];

  float* nodes_out = (float*)d_out;
  float* edges_out = (float*)d_out + (size_t)nNodes * H;
  float* recv_agg  = (float*)d_ws;  // N*H fp32 = 25.6 MB, fits in 192 MB L2

  zero_kernel<<<2048, 256, 0, stream>>>(recv_agg, (size_t)nNodes * H);

  const int eTiles = (nEdges + 15) / 16;
  const int eBlocks = (eTiles + 7) / 8;
  const size_t eShm = (size_t)(192 * H + 8 * 16 * H) * sizeof(float);  // 80 KB
  edge_kernel<<<eBlocks, 256, eShm, stream>>>(nodes, edges, senders, receivers,
                                              We, be, ln_e_scale, ln_e_bias,
                                              edges_out, recv_agg, nEdges);

  const int nTiles = (nNodes + 15) / 16;
  const int nBlocks = (nTiles + 7) / 8;
  const size_t nShm = (size_t)(128 * H + 8 * 16 * H) * sizeof(float);  // 64 KB
  node_kernel<<<nBlocks, 256, nShm, stream>>>(nodes, recv_agg, Wn, bn,
                                              ln_n_scale, ln_n_bias, nodes_out,
                                              nNodes);
}